// FLDD_30434138260220
// MI455X (gfx1250) — compile-verified
//
#include <hip/hip_runtime.h>

typedef float v2f __attribute__((ext_vector_type(2)));
typedef float v8f __attribute__((ext_vector_type(8)));
typedef unsigned int u32x4 __attribute__((ext_vector_type(4)));
typedef int i32x4 __attribute__((ext_vector_type(4)));
typedef int i32x8 __attribute__((ext_vector_type(8)));

static constexpr int   KC   = 64;     // K classes
static constexpr int   LSEQ = 2048;   // L
static constexpr float EPSF = 1e-6f;

#define LOG_EPS_F   (-13.815510557964274f)   /* log(1e-6)   */
#define LOG1P_EPS_F (9.999995000003333e-07f) /* log1p(1e-6) */
#define LOG_UNIF_F  (-4.1588830833596715f)   /* log(1/64)   */

#if defined(__HIP_DEVICE_COMPILE__) && \
    __has_builtin(__builtin_amdgcn_tensor_load_to_lds) && \
    __has_builtin(__builtin_amdgcn_s_wait_tensorcnt)
#define USE_TDM 1
#else
#define USE_TDM 0
#endif

__device__ __forceinline__ float lae(float a, float b) {   // logaddexp
    float m = fmaxf(a, b);
    return m + log1pf(expf(-fabsf(a - b)));
}
__device__ __forceinline__ float red_add16(float v) {      // sum over 16-lane half
    v += __shfl_xor(v, 1, 32);
    v += __shfl_xor(v, 2, 32);
    v += __shfl_xor(v, 4, 32);
    v += __shfl_xor(v, 8, 32);
    return v;
}
__device__ __forceinline__ float red_max16(float v) {      // max over 16-lane half
    v = fmaxf(v, __shfl_xor(v, 1, 32));
    v = fmaxf(v, __shfl_xor(v, 2, 32));
    v = fmaxf(v, __shfl_xor(v, 4, 32));
    v = fmaxf(v, __shfl_xor(v, 8, 32));
    return v;
}
// control_lp = log_linear_mix(one_hot_logprob, log(1/K), w): only 2 distinct values
__device__ __forceinline__ void control_hm(float w, float& hit, float& miss) {
    if (w < 1e-6f)             { hit = LOG1P_EPS_F; miss = LOG_EPS_F; }
    else if (w > 1.f - 1e-6f)  { hit = LOG_UNIF_F;  miss = LOG_UNIF_F; }
    else {
        float lw = logf(w), l1w = logf(1.f - w);
        hit  = lae(LOG1P_EPS_F + l1w, LOG_UNIF_F + lw);
        miss = lae(LOG_EPS_F   + l1w, LOG_UNIF_F + lw);
    }
}
__device__ __forceinline__ float mix2(float a, float cl, int mode, float l1r, float lr) {
    if (mode == 0) return a;        // w < 1e-6
    if (mode == 1) return cl;       // w > 1-1e-6
    return lae(a + l1r, cl + lr);
}

__global__ __launch_bounds__(256, 1) void fldd_kernel(
    const float* __restrict__ W_emb,   // (64,64)
    const float* __restrict__ W_t,     // (1001,64)
    const int*   __restrict__ x0g,     // (B,L)
    const int*   __restrict__ xtg,     // (B,L)
    const int*   __restrict__ tg,      // (B,)
    float*       __restrict__ out)     // (B,L,64)
{
    __shared__ float lds_w[KC * KC];   // 16 KB: W_emb row-major, LDS offset 0

    const int tid  = threadIdx.x;
    const int lane = tid & 31;
    const int wave = tid >> 5;
    const int myn  = lane & 15;        // N-column within 16-wide tile
    const int hi   = lane >> 4;        // 0 | 1 (lane half)

#if USE_TDM
    // ---- Stage W_emb into LDS with the Tensor Data Mover (one DMA / block) --
    if (wave == 0) {
        const unsigned long long ga = (unsigned long long)(uintptr_t)W_emb;
        // Flat shared pointer: low 32 bits == LDS byte address (aperture rule).
        const unsigned ldsoff = (unsigned)(uintptr_t)(void*)lds_w;
        u32x4 g0;
        g0[0] = 1u;                                        // count=1, is_restore=0
        g0[1] = ldsoff;                                    // lds_addr
        g0[2] = (unsigned)(ga & 0xFFFFFFFFu);              // global_addr[31:0]
        g0[3] = (unsigned)((ga >> 32) & 0x01FFFFFFu)       // global_addr[56:32]
              | 0x80000000u;                               // type=2 ("image")
        i32x8 g1;
        g1[0] = (int)0x00020000;   // wg_mask=0, data_size=2 (4 B), no pad/iter
        g1[1] = (int)(64u << 16);  // tensor_dim0 = 64 (bits 79:48, low half)
        g1[2] = (int)(64u << 16);  // tensor_dim1 = 64 (bits 111:80, low half)
        g1[3] = (int)(64u << 16);  // tile_dim0   = 64 (bits 127:112)
        g1[4] = 64;                // tile_dim1   = 64 (bits 143:128)
        g1[5] = 64;                // tensor_dim0_stride = 64 (bits 207:160)
        g1[6] = 0;
        g1[7] = 0;                 // tensor_dim1_stride = 0 (2-D)
        i32x4 z4 = {0, 0, 0, 0};   // D# groups 2/3 unused (2-D tensor)
#if defined(__clang_major__) && (__clang_major__ >= 23)
        i32x8 z8 = {0, 0, 0, 0, 0, 0, 0, 0};
        __builtin_amdgcn_tensor_load_to_lds(g0, g1, z4, z4, z8, 0);
#else
        __builtin_amdgcn_tensor_load_to_lds(g0, g1, z4, z4, 0);
#endif
    }
#else
    // ---- Fallback: cooperative staging (256 threads x 4 float4) ------------
    {
        const float4* src = reinterpret_cast<const float4*>(W_emb);
        float4*       dst = reinterpret_cast<float4*>(lds_w);
        #pragma unroll
        for (int i = 0; i < 4; ++i)
            dst[tid + i * 256] = src[tid + i * 256];
    }
#endif

    // One wave = 16 tokens; 8 waves/block; tiles never straddle b (2048 % 16 == 0)
    const int tile = blockIdx.x * 8 + wave;
    const int base = tile * 16;
    const int b    = base >> 11;       // / LSEQ

    const int v_x0 = x0g[base + myn];  // token (base+myn) metadata, per lane
    const int v_xt = xtg[base + myn];
    const int tb   = tg[b];            // uniform

    __builtin_prefetch(W_t + (size_t)(tb - 1) * KC, 0, 0);

    // W_t rows for t and t-1 at class col = nt*16 + myn (r-independent)
    float wtt[4], wts[4];
    #pragma unroll
    for (int nt = 0; nt < 4; ++nt) {
        wtt[nt] = W_t[(size_t)tb * KC + nt * 16 + myn];
        wts[nt] = W_t[(size_t)(tb - 1) * KC + nt * 16 + myn];
    }

    // per-r token metadata: token m = r + 8*hi lives on source lane m (m < 16)
    int x0r[8], jr[8];
    #pragma unroll
    for (int r = 0; r < 8; ++r) {
        const int m = r + 8 * hi;
        x0r[r] = __shfl(v_x0, m, 32);
        jr[r]  = __shfl(v_xt, m, 32);
    }

#if USE_TDM
    if (wave == 0) __builtin_amdgcn_s_wait_tensorcnt(0);   // DMA done before barrier
#endif
    __syncthreads();

    // ---- Gather G[m][k] = W_emb[x_0[m]][k] as one_hot(x0) @ W_emb ----------
    // C/D layout: vgpr r -> (M = r + 8*hi, N = nt*16 + myn)
    v8f zero = {0.f,0.f,0.f,0.f,0.f,0.f,0.f,0.f};
    v8f acc[4] = {zero, zero, zero, zero};
#if __has_builtin(__builtin_amdgcn_wmma_f32_16x16x4_f32)
    #pragma unroll
    for (int nt = 0; nt < 4; ++nt) {
        const int col = nt * 16 + myn;
        #pragma unroll
        for (int c0 = 0; c0 < KC; c0 += 4) {
            const int c = c0 + 2 * hi;                 // A/B 16x4 lane-half split
            v2f a;                                     // exact 0/1 one-hot rows
            a.x = (v_x0 == c    ) ? 1.f : 0.f;
            a.y = (v_x0 == c + 1) ? 1.f : 0.f;
            v2f bm;                                    // W_emb[c..c+1][col]
            bm.x = lds_w[ c      * KC + col];
            bm.y = lds_w[(c + 1) * KC + col];
            acc[nt] = __builtin_amdgcn_wmma_f32_16x16x4_f32(
                false, a, false, bm, (short)0, acc[nt], false, false);
        }
    }
#else
    #pragma unroll
    for (int nt = 0; nt < 4; ++nt) {
        const int col = nt * 16 + myn;
        #pragma unroll
        for (int r = 0; r < 8; ++r)
            acc[nt][r] = lds_w[x0r[r] * KC + col];
    }
#endif

    // ---- per-b mixing scalars ----------------------------------------------
    const float wt = (float)tb / 1000.0f;
    const float ws = (float)(tb - 1) / 1000.0f;
    float rt = 2.f * wt - 1.f; rt *= rt;               // control_ratio_t
    float rs = 2.f * ws - 1.f; rs *= rs;

    float clt_hit, clt_miss, cls_hit, cls_miss;
    control_hm(wt, clt_hit, clt_miss);
    control_hm(ws, cls_hit, cls_miss);

    int mode_t; float lrt = 0.f, l1rt = 0.f;
    if (rt < 1e-6f)            mode_t = 0;
    else if (rt > 1.f - 1e-6f) mode_t = 1;
    else { mode_t = 2; lrt = logf(rt); l1rt = logf(1.f - rt); }
    int mode_s; float lrs = 0.f, l1rs = 0.f;
    if (rs < 1e-6f)            mode_s = 0;
    else if (rs > 1.f - 1e-6f) mode_s = 1;
    else { mode_s = 2; lrs = logf(rs); l1rs = logf(1.f - rs); }

    // ---- raw logits --------------------------------------------------------
    float at[4][8], as_[4][8];
    #pragma unroll
    for (int nt = 0; nt < 4; ++nt)
        #pragma unroll
        for (int r = 0; r < 8; ++r) {
            float g = acc[nt][r];
            at[nt][r]  = g + wtt[nt];
            as_[nt][r] = g + wts[nt];
        }

    // ---- log_softmax over K (per token r) ----------------------------------
    #pragma unroll
    for (int r = 0; r < 8; ++r) {
        float mt = red_max16(fmaxf(fmaxf(at[0][r],  at[1][r]),  fmaxf(at[2][r],  at[3][r])));
        float ms = red_max16(fmaxf(fmaxf(as_[0][r], as_[1][r]), fmaxf(as_[2][r], as_[3][r])));
        float et = expf(at[0][r]-mt)  + expf(at[1][r]-mt)  + expf(at[2][r]-mt)  + expf(at[3][r]-mt);
        float es = expf(as_[0][r]-ms) + expf(as_[1][r]-ms) + expf(as_[2][r]-ms) + expf(as_[3][r]-ms);
        float lset = mt + logf(red_add16(et));
        float lses = ms + logf(red_add16(es));
        #pragma unroll
        for (int nt = 0; nt < 4; ++nt) { at[nt][r] -= lset; as_[nt][r] -= lses; }
    }

    // ---- mix with control distribution ------------------------------------
    #pragma unroll
    for (int nt = 0; nt < 4; ++nt) {
        const int col = nt * 16 + myn;
        #pragma unroll
        for (int r = 0; r < 8; ++r) {
            const bool hitc = (col == x0r[r]);
            at[nt][r]  = mix2(at[nt][r],  hitc ? clt_hit : clt_miss, mode_t, l1rt, lrt);
            as_[nt][r] = mix2(as_[nt][r], hitc ? cls_hit : cls_miss, mode_s, l1rs, lrs);
        }
    }

    // ---- softmax -> u_t, u_s (stored back into at / as_) -------------------
    #pragma unroll
    for (int r = 0; r < 8; ++r) {
        float mt = red_max16(fmaxf(fmaxf(at[0][r],  at[1][r]),  fmaxf(at[2][r],  at[3][r])));
        float ms = red_max16(fmaxf(fmaxf(as_[0][r], as_[1][r]), fmaxf(as_[2][r], as_[3][r])));
        #pragma unroll
        for (int nt = 0; nt < 4; ++nt) {
            at[nt][r]  = expf(at[nt][r]  - mt);
            as_[nt][r] = expf(as_[nt][r] - ms);
        }
        float Zt = red_add16(at[0][r]  + at[1][r]  + at[2][r]  + at[3][r]);
        float Zs = red_add16(as_[0][r] + as_[1][r] + as_[2][r] + as_[3][r]);
        #pragma unroll
        for (int nt = 0; nt < 4; ++nt) { at[nt][r] /= Zt; as_[nt][r] /= Zs; }
    }

    // ---- maximal coupling, row j = x_t only --------------------------------
    #pragma unroll
    for (int r = 0; r < 8; ++r) {
        float pd = 0.f, put = 0.f, pus = 0.f;
        #pragma unroll
        for (int nt = 0; nt < 4; ++nt) {
            const int col = nt * 16 + myn;
            pd += fmaxf(as_[nt][r] - at[nt][r], 0.f);     // relu(u_s - u_t)
            if (col == jr[r]) { put += at[nt][r]; pus += as_[nt][r]; }
        }
        const float S   = red_add16(pd);
        const float utj = red_add16(put);                 // u_t[j]
        const float usj = red_add16(pus);                 // u_s[j]

        const float pstay  = fminf(usj, utj) / (utj + EPSF);
        const float pmove  = fmaxf(utj - usj, 0.f) / (utj + EPSF);
        const float rowsum = pstay + pmove * (S / (S + EPSF));

        const int token = base + r + 8 * hi;
        #pragma unroll
        for (int nt = 0; nt < 4; ++nt) {
            const int col = nt * 16 + myn;
            float mst = fmaxf(as_[nt][r] - at[nt][r], 0.f) / (S + EPSF);
            float q   = pmove * mst + ((col == jr[r]) ? pstay : 0.f);
            out[(size_t)token * KC + col] = logf(q / (rowsum + EPSF) + EPSF);
        }
    }
}

extern "C" void kernel_launch(void* const* d_in, const int* in_sizes, int n_in,
                              void* d_out, int out_size, void* d_ws, size_t ws_size,
                              hipStream_t stream) {
    (void)in_sizes; (void)n_in; (void)out_size; (void)d_ws; (void)ws_size;
    const float* W_emb = (const float*)d_in[0];
    const float* W_t   = (const float*)d_in[1];
    const int*   x0    = (const int*)d_in[2];
    const int*   xt    = (const int*)d_in[3];
    const int*   t     = (const int*)d_in[4];
    float* out = (float*)d_out;

    const int tokens = 4 * LSEQ;          // B*L = 8192
    const int tiles  = tokens / 16;       // 512 waves
    dim3 block(256);                      // 8 waves/block
    dim3 grid(tiles / 8);                 // 64 blocks
    fldd_kernel<<<grid, block, 0, stream>>>(W_emb, W_t, x0, xt, t, out);
}